// SPDNetwork_43997644980738
// MI455X (gfx1250) — compile-verified
//
#include <hip/hip_runtime.h>

typedef float v2f __attribute__((ext_vector_type(2)));
typedef float v4f __attribute__((ext_vector_type(4)));
typedef float v8f __attribute__((ext_vector_type(8)));
typedef unsigned int u32x4 __attribute__((ext_vector_type(4)));
typedef int i32x4 __attribute__((ext_vector_type(4)));
typedef int i32x8 __attribute__((ext_vector_type(8)));

#define REEIG_EPS 1e-4f
#define XC_STRIDE 132   // 128 + 4 pad: 16B alignment for b128/b64 LDS ops, bank skew
                        // (reproduced by TDM pad_interval=6 [128 dw], pad_amount=3 [4 dw])

// ---------------------------------------------------------------------------
// Parallel cyclic Jacobi eigensolver + eigenvalue rectification + reconstruct.
// A,V,R are N*N LDS buffers. One wave32 works cooperatively; N/2 disjoint
// rotations are applied per round (circle/round-robin pairing), N-1 rounds per
// sweep. Single-wave workgroup => __syncthreads() lowers to a cheap barrier.
// ---------------------------------------------------------------------------
template<int N, int NSWEEP>
__device__ __forceinline__ void reeig_jacobi(
    float* __restrict__ A, float* __restrict__ V, float* __restrict__ R,
    int* __restrict__ part, float* __restrict__ colC, float* __restrict__ colS,
    int lane)
{
  constexpr int NP = N / 2;            // disjoint pairs per round
  constexpr int NR = N - 1;            // rounds per sweep
  constexpr int ITEMS = (N * N + 31) / 32;

  for (int idx = lane; idx < N * N; idx += 32)
    V[idx] = ((idx / N) == (idx % N)) ? 1.0f : 0.0f;
  __syncthreads();

  for (int sweep = 0; sweep < NSWEEP; ++sweep) {
    for (int r = 0; r < NR; ++r) {
      // one lane per pair computes its Givens rotation
      if (lane < NP) {
        int p, q;
        if (lane == 0) { p = N - 1; q = r; }
        else {
          p = (r + lane) % (N - 1);
          q = (r - lane + (N - 1)) % (N - 1);
        }
        if (p > q) { int t = p; p = q; q = t; }
        float app = A[p * N + p], aqq = A[q * N + q], apq = A[p * N + q];
        float cc = 1.0f, ss = 0.0f;
        if (fabsf(apq) > 1e-12f) {
          float tau = (aqq - app) / (2.0f * apq);
          float t = (tau >= 0.0f ? 1.0f : -1.0f) /
                    (fabsf(tau) + sqrtf(1.0f + tau * tau));
          cc = rsqrtf(1.0f + t * t);
          ss = t * cc;
        }
        part[p] = q;  part[q] = p;
        colC[p] = cc; colC[q] = cc;
        colS[p] = -ss; colS[q] = ss;   // J[p][p]=c, J[q][p]=-s, J[p][q]=s, J[q][q]=c
      }
      __syncthreads();

      float ra[ITEMS], rb[ITEMS];

      // A <- A * J  (mix columns; all pairs disjoint, applied at once)
      #pragma unroll
      for (int t = 0; t < ITEMS; ++t) {
        int idx = lane + 32 * t;
        if (idx < N * N) {
          int i = idx / N, c = idx % N;
          ra[t] = A[i * N + c];
          rb[t] = A[i * N + part[c]];
        }
      }
      __syncthreads();
      #pragma unroll
      for (int t = 0; t < ITEMS; ++t) {
        int idx = lane + 32 * t;
        if (idx < N * N) {
          int c = idx % N;
          A[idx] = colC[c] * ra[t] + colS[c] * rb[t];
        }
      }
      __syncthreads();

      // A <- J^T * A (mix rows; same coefficient arrays by symmetry)
      #pragma unroll
      for (int t = 0; t < ITEMS; ++t) {
        int idx = lane + 32 * t;
        if (idx < N * N) {
          int i = idx / N, j = idx % N;
          ra[t] = A[i * N + j];
          rb[t] = A[part[i] * N + j];
        }
      }
      __syncthreads();
      #pragma unroll
      for (int t = 0; t < ITEMS; ++t) {
        int idx = lane + 32 * t;
        if (idx < N * N) {
          int i = idx / N;
          A[idx] = colC[i] * ra[t] + colS[i] * rb[t];
        }
      }
      __syncthreads();

      // V <- V * J (accumulate eigenvectors)
      #pragma unroll
      for (int t = 0; t < ITEMS; ++t) {
        int idx = lane + 32 * t;
        if (idx < N * N) {
          int i = idx / N, c = idx % N;
          ra[t] = V[i * N + c];
          rb[t] = V[i * N + part[c]];
        }
      }
      __syncthreads();
      #pragma unroll
      for (int t = 0; t < ITEMS; ++t) {
        int idx = lane + 32 * t;
        if (idx < N * N) {
          int c = idx % N;
          V[idx] = colC[c] * ra[t] + colS[c] * rb[t];
        }
      }
      __syncthreads();
    }
  }

  // R = V * max(diag(A), eps) * V^T   (eigenvalue rectification)
  for (int idx = lane; idx < N * N; idx += 32) {
    int i = idx / N, j = idx % N;
    float acc = 0.0f;
    for (int k = 0; k < N; ++k)
      acc += V[i * N + k] * fmaxf(A[k * N + k], REEIG_EPS) * V[j * N + k];
    R[idx] = acc;
  }
  __syncthreads();
}

// ---------------------------------------------------------------------------
// Fully fused SPDNet forward: one wave32 per batch element.
// x tile staged by the Tensor Data Mover (async DMA global->LDS, TENSORcnt).
// ---------------------------------------------------------------------------
__global__ __launch_bounds__(32)
void spdnet_fused_kernel(const float* __restrict__ x,
                         const float* __restrict__ w1,
                         const float* __restrict__ w2,
                         const float* __restrict__ w3,
                         const float* __restrict__ wfc,
                         float* __restrict__ out)
{
  __shared__ float xc[32 * XC_STRIDE];     // centered 32x128 tile (TDM-padded rows)
  __shared__ float cov[32 * 32];           // covariance / scratch
  __shared__ float tmp[32 * 16];           // bimap intermediate
  __shared__ float Abuf[16 * 16], Vbuf[16 * 16], Rbuf[16 * 16];
  __shared__ float w1s[32 * 16], w2s[16 * 8], w3s[8 * 4], wfcs[16 * 10];
  __shared__ int   part[16];
  __shared__ float colC[16], colS[16];

  const int lane = threadIdx.x;
  const int b    = blockIdx.x;
  const float* __restrict__ xb = x + (size_t)b * 32 * 128;
  const float inv32 = 1.0f / 32.0f;

  // ---- kick off TDM: DMA the 32x128 f32 tile into LDS with 4-dword row pad.
  //      D# group0: count=1, lds_addr, 57b global_addr, type=2 ("image").
  //      D# group1: data_size=4B, pad_enable, pad_interval=128dw, pad_amount=4dw,
  //                 tensor_dim0=128, tensor_dim1=32, tile=128x32, stride0=128.
  {
    unsigned int lds_off = (unsigned int)(uintptr_t)&xc[0];   // low 32b = LDS offset
    unsigned long long ga = (unsigned long long)(uintptr_t)xb;
    u32x4 g0 = { 1u,                                   // count=1, user mode
                 lds_off,
                 (unsigned int)(ga & 0xFFFFFFFFu),
                 (unsigned int)((ga >> 32) & 0x01FFFFFFu) | (2u << 30) };
    i32x8 g1 = { (int)0x07920000u,                     // dsz=4B, pad_en, int=6, amt=3
                 (int)(128u << 16),                    // tensor_dim0[15:0]=128
                 (int)(32u << 16),                     // dim0[31:16]=0 | tensor_dim1[15:0]=32
                 (int)(128u << 16),                    // dim1[31:16]=0 | tile_dim0=128
                 (int)32,                              // tile_dim1=32, tile_dim2=0
                 (int)128,                             // tensor_dim0_stride=128
                 (int)(0x1000u << 16),                 // tensor_dim1_stride[15:0]=4096
                 0 };                                  // tensor_dim1_stride[47:16]=0
    i32x4 g2 = { 0, 0, 0, 0 };
    i32x4 g3 = { 0, 0, 0, 0 };
#if __clang_major__ >= 23
    i32x8 gx = { 0, 0, 0, 0, 0, 0, 0, 0 };
    __builtin_amdgcn_tensor_load_to_lds(g0, g1, g2, g3, gx, 0);
#else
    __builtin_amdgcn_tensor_load_to_lds(g0, g1, g2, g3, 0);
#endif
  }

  // stage weights while the DMA is in flight (tiny, broadcast-friendly)
  for (int i = lane; i < 512; i += 32) w1s[i] = w1[i];
  for (int i = lane; i < 128; i += 32) w2s[i] = w2[i];
  w3s[lane] = w3[lane];
  for (int i = lane; i < 160; i += 32) wfcs[i] = wfc[i];

  __builtin_amdgcn_s_wait_tensorcnt(0);    // tile resident in LDS
  __syncthreads();

  // ---- center: mean over the CHANNEL dim per time point (lane owns 4 cols) ----
  float s0 = 0.f, s1 = 0.f, s2 = 0.f, s3 = 0.f;
  for (int c = 0; c < 32; ++c) {
    v4f v = *(const v4f*)(&xc[c * XC_STRIDE + lane * 4]);
    s0 += v.x; s1 += v.y; s2 += v.z; s3 += v.w;
  }
  s0 *= inv32; s1 *= inv32; s2 *= inv32; s3 *= inv32;
  for (int c = 0; c < 32; ++c) {
    v4f v = *(v4f*)(&xc[c * XC_STRIDE + lane * 4]);
    v.x -= s0; v.y -= s1; v.z -= s2; v.w -= s3;
    *(v4f*)(&xc[c * XC_STRIDE + lane * 4]) = v;
  }
  __syncthreads();

  // ---- cov = xc * xc^T / 32 on the f32 WMMA path ----
  // A fragment (16x4 f32): lanes 0-15 row M=l, K=k..k+1; lanes 16-31 row M=l,
  // K=k+2..k+3. For B = xc^T the per-lane fragment is *identical*, so each
  // K-step is two ds_load_b64 + four v_wmma_f32_16x16x4_f32.
  const int half = lane >> 4;
  const int l16  = lane & 15;
  v8f c00 = {}, c01 = {}, c10 = {}, c11 = {};
  #pragma unroll 4
  for (int k = 0; k < 128; k += 4) {
    v2f aLo = *(const v2f*)(&xc[ l16       * XC_STRIDE + k + 2 * half]);
    v2f aHi = *(const v2f*)(&xc[(l16 + 16) * XC_STRIDE + k + 2 * half]);
    c00 = __builtin_amdgcn_wmma_f32_16x16x4_f32(false, aLo, false, aLo, (short)0, c00, false, false);
    c01 = __builtin_amdgcn_wmma_f32_16x16x4_f32(false, aLo, false, aHi, (short)0, c01, false, false);
    c10 = __builtin_amdgcn_wmma_f32_16x16x4_f32(false, aHi, false, aLo, (short)0, c10, false, false);
    c11 = __builtin_amdgcn_wmma_f32_16x16x4_f32(false, aHi, false, aHi, (short)0, c11, false, false);
  }
  // spill accumulators: VGPR r holds row r (lanes 0-15) / row r+8 (lanes 16-31)
  #pragma unroll
  for (int r = 0; r < 8; ++r) {
    int row = r + half * 8;
    cov[ row       * 32 + l16     ] = c00[r] * inv32;
    cov[ row       * 32 + l16 + 16] = c01[r] * inv32;
    cov[(row + 16) * 32 + l16     ] = c10[r] * inv32;
    cov[(row + 16) * 32 + l16 + 16] = c11[r] * inv32;
  }
  __syncthreads();

  // +0.01*trace added as a SCALAR to every entry (per reference)
  float trv = 0.0f;
  for (int i = 0; i < 32; ++i) trv += cov[i * 32 + i];
  float tadd = 0.01f * trv;
  for (int i = lane; i < 1024; i += 32) cov[i] += tadd;
  __syncthreads();

  // ---- BiMap 1: s1 = w1^T * cov * w1  (32->16) ----
  {
    int j = l16;
    for (int i0 = 0; i0 < 16; ++i0) {          // tmp = cov * w1   [32x16]
      int i = i0 + 16 * half;
      float acc = 0.0f;
      #pragma unroll 8
      for (int p = 0; p < 32; ++p) acc += cov[i * 32 + p] * w1s[p * 16 + j];
      tmp[i * 16 + j] = acc;
    }
    __syncthreads();
    for (int i0 = 0; i0 < 8; ++i0) {           // s1 = w1^T * tmp  [16x16]
      int i = i0 + 8 * half;
      float acc = 0.0f;
      #pragma unroll 8
      for (int p = 0; p < 32; ++p) acc += w1s[p * 16 + i] * tmp[p * 16 + j];
      Rbuf[i * 16 + j] = acc;
    }
  }
  __syncthreads();
  for (int idx = lane; idx < 256; idx += 32) {  // symmetrize
    int i = idx >> 4, j = idx & 15;
    Abuf[idx] = 0.5f * (Rbuf[i * 16 + j] + Rbuf[j * 16 + i]);
  }
  __syncthreads();
  reeig_jacobi<16, 8>(Abuf, Vbuf, Rbuf, part, colC, colS, lane);   // r1 in Rbuf

  // ---- BiMap 2: s2 = w2^T * r1 * w2  (16->8) ----
  for (int idx = lane; idx < 128; idx += 32) { // tmp2 = r1 * w2  [16x8]
    int p = idx >> 3, j = idx & 7;
    float acc = 0.0f;
    #pragma unroll
    for (int t = 0; t < 16; ++t) acc += Rbuf[p * 16 + t] * w2s[t * 8 + j];
    tmp[idx] = acc;
  }
  __syncthreads();
  for (int idx = lane; idx < 64; idx += 32) {  // s2 = w2^T * tmp2 [8x8]
    int i = idx >> 3, j = idx & 7;
    float acc = 0.0f;
    #pragma unroll
    for (int p = 0; p < 16; ++p) acc += w2s[p * 8 + i] * tmp[p * 8 + j];
    cov[idx] = acc;
  }
  __syncthreads();
  for (int idx = lane; idx < 64; idx += 32) {  // symmetrize
    int i = idx >> 3, j = idx & 7;
    Abuf[idx] = 0.5f * (cov[i * 8 + j] + cov[j * 8 + i]);
  }
  __syncthreads();
  reeig_jacobi<8, 8>(Abuf, Vbuf, Rbuf, part, colC, colS, lane);    // r2 in Rbuf

  // ---- BiMap 3: s3 = w3^T * r2 * w3  (8->4) ----
  {                                            // tmp3 = r2 * w3  [8x4]
    int p = lane >> 2, j = lane & 3;
    float acc = 0.0f;
    #pragma unroll
    for (int t = 0; t < 8; ++t) acc += Rbuf[p * 8 + t] * w3s[t * 4 + j];
    tmp[lane] = acc;
  }
  __syncthreads();
  if (lane < 16) {                             // s3 = w3^T * tmp3 [4x4]
    int i = lane >> 2, j = lane & 3;
    float acc = 0.0f;
    #pragma unroll
    for (int p = 0; p < 8; ++p) acc += w3s[p * 4 + i] * tmp[p * 4 + j];
    cov[lane] = acc;
  }
  __syncthreads();
  if (lane < 16) {
    int i = lane >> 2, j = lane & 3;
    Abuf[lane] = 0.5f * (cov[i * 4 + j] + cov[j * 4 + i]);
  }
  __syncthreads();
  reeig_jacobi<4, 6>(Abuf, Vbuf, Rbuf, part, colC, colS, lane);    // r3 in Rbuf (16 floats)

  // ---- FC head: logits[o] = sum_k r3.flat[k] * wfc[k][o] ----
  if (lane < 10) {
    float acc = 0.0f;
    #pragma unroll
    for (int k = 0; k < 16; ++k) acc += Rbuf[k] * wfcs[k * 10 + lane];
    out[(size_t)b * 10 + lane] = acc;
  }
}

extern "C" void kernel_launch(void* const* d_in, const int* in_sizes, int n_in,
                              void* d_out, int out_size, void* d_ws, size_t ws_size,
                              hipStream_t stream) {
  (void)n_in; (void)out_size; (void)d_ws; (void)ws_size;
  const float* x   = (const float*)d_in[0];   // [B,1,32,128]
  const float* w1  = (const float*)d_in[1];   // [1,32,16]
  const float* w2  = (const float*)d_in[2];   // [1,16,8]
  const float* w3  = (const float*)d_in[3];   // [1,8,4]
  const float* wfc = (const float*)d_in[4];   // [1,16,10]
  float* out = (float*)d_out;                 // [1,B,10]
  const int B = in_sizes[0] / (32 * 128);
  spdnet_fused_kernel<<<B, 32, 0, stream>>>(x, w1, w2, w3, wfc, out);
}